// InterViews_35622458753792
// MI455X (gfx1250) — compile-verified
//
#include <hip/hip_runtime.h>

typedef __attribute__((ext_vector_type(2))) float v2f;
typedef __attribute__((ext_vector_type(4))) float v4f;
typedef __attribute__((ext_vector_type(8))) float v8f;

#define NV 16      // views
#define NB 1024    // batch
#define NC 4096    // channels
#define WAVES_PER_BLOCK 8
#define BATCHES_PER_BLOCK 4   // 2 waves (split-K halves) per batch

__global__ __launch_bounds__(256)
void interviews_gramvar_kernel(const float* __restrict__ vf, float* __restrict__ out) {
    // partial Gram from the odd (K-upper-half) wave: 32 lanes x 8 floats per batch slot
    __shared__ float s_part[BATCHES_PER_BLOCK][32 * 8];
    __shared__ float s_diag[BATCHES_PER_BLOCK][NV];

    const int lane = threadIdx.x & 31;
    const int wave = threadIdx.x >> 5;
    const int pair = wave >> 1;        // batch slot within block
    const int half = wave & 1;         // which K half this wave reduces
    const int b    = blockIdx.x * BATCHES_PER_BLOCK + pair;

    const int r  = lane & 15;          // view/row this lane feeds (A and B identical for Gram)
    const int hi = lane >> 4;          // half-wave: selects K slots {0,1} vs {2,3}

    // row r of batch b: vf[(r*NB + b)*NC + c]
    const float* rowp = vf + ((size_t)r * NB + (size_t)b) * NC;
    const int kbeg = half * (NC / 2);
    const int kend = kbeg + (NC / 2);

    v8f acc = {0.f, 0.f, 0.f, 0.f, 0.f, 0.f, 0.f, 0.f};

    // Each iteration covers 16 columns with 2 coalesced b128 loads + 4 f32 WMMAs.
    // WMMA K-slot mapping (lane half hi, vgpr j) -> logical k = 2*hi + j; any consistent
    // permutation of actual columns onto logical k is valid since A == B.
    #pragma unroll 4
    for (int k0 = kbeg; k0 < kend; k0 += 16) {
        const v4f p = *(const v4f*)(rowp + k0 + 4 * hi);
        const v4f q = *(const v4f*)(rowp + k0 + 8 + 4 * hi);
        v2f a0 = {p.x, p.y};
        v2f a1 = {p.z, p.w};
        v2f a2 = {q.x, q.y};
        v2f a3 = {q.z, q.w};
        acc = __builtin_amdgcn_wmma_f32_16x16x4_f32(false, a0, false, a0, (short)0, acc, false, false);
        acc = __builtin_amdgcn_wmma_f32_16x16x4_f32(false, a1, false, a1, (short)0, acc, false, false);
        acc = __builtin_amdgcn_wmma_f32_16x16x4_f32(false, a2, false, a2, (short)0, acc, false, false);
        acc = __builtin_amdgcn_wmma_f32_16x16x4_f32(false, a3, false, a3, (short)0, acc, false, false);
    }

    // ---- combine split-K halves (identical accumulator layouts -> per-lane add) ----
    if (half == 1) {
        #pragma unroll
        for (int j = 0; j < 8; ++j) s_part[pair][lane * 8 + j] = acc[j];
    }
    __syncthreads();

    if (half == 0) {
        #pragma unroll
        for (int j = 0; j < 8; ++j) acc[j] += s_part[pair][lane * 8 + j];

        // C/D layout: vgpr j, lane -> G[M][N], M = j + 8*hi, N = r.
        // Diagonal element lives where j + 8*hi == r.
        float dv = 0.f;
        #pragma unroll
        for (int j = 0; j < 8; ++j)
            if (j + 8 * hi == r) dv = acc[j];
        if ((hi == 0 && r < 8) || (hi == 1 && r >= 8)) s_diag[pair][r] = dv;
    }
    __syncthreads();

    // turn diagonal into 1/norm (16 lanes per batch do it)
    if (half == 0 && hi == 0) s_diag[pair][r] = 1.0f / sqrtf(s_diag[pair][r]);
    __syncthreads();

    if (half == 0) {
        const float invN = s_diag[pair][r];
        float sum = 0.f, sumsq = 0.f;
        #pragma unroll
        for (int j = 0; j < 8; ++j) {
            const int M = j + 8 * hi;
            const float s = acc[j] * s_diag[pair][M] * invN;
            if (M < r) {           // strict upper triangle: each unordered pair once (120 values)
                sum   += s;
                sumsq += s * s;
            }
        }
        // wave32 butterfly reduction
        #pragma unroll
        for (int off = 16; off > 0; off >>= 1) {
            sum   += __shfl_xor(sum,   off, 32);
            sumsq += __shfl_xor(sumsq, off, 32);
        }
        if (lane == 0) {
            const float n   = 120.0f;
            const float var = (sumsq - sum * sum / n) / (n - 1.0f);
            out[b] = -var;
        }
    }
}

extern "C" void kernel_launch(void* const* d_in, const int* in_sizes, int n_in,
                              void* d_out, int out_size, void* d_ws, size_t ws_size,
                              hipStream_t stream) {
    const float* vf  = (const float*)d_in[0];   // (16*1024, 4096) fp32
    float*       out = (float*)d_out;           // 1024 fp32
    dim3 grid(NB / BATCHES_PER_BLOCK);          // 256 blocks
    dim3 block(32 * WAVES_PER_BLOCK);           // 256 threads = 8 waves
    interviews_gramvar_kernel<<<grid, block, 0, stream>>>(vf, out);
}